// MaxMarginLoss_29360396435938
// MI455X (gfx1250) — compile-verified
//
#include <hip/hip_runtime.h>
#include <hip/hip_bf16.h>

// Problem constants (match reference)
#define BB 16
#define TT 2048
#define DD 1024
#define SS 32
#define KT 128         // timesteps staged per LDS chunk (32 KB / buffer)
#define NC (TT / KT)   // 16 chunks
#define ALPHA_C 1.0f

typedef float v2f __attribute__((ext_vector_type(2)));
typedef float v8f __attribute__((ext_vector_type(8)));
typedef unsigned int u32x4 __attribute__((ext_vector_type(4)));
typedef unsigned int u32x8 __attribute__((ext_vector_type(8)));

// ---------------------------------------------------------------------------
// Tensor Data Mover: DMA a KT x 64-float tile of x (row stride D) into LDS.
// Descriptor per CDNA5 ISA ch.8 (D# groups 0/1, VADDR2/3 = NULL -> 2D tensor).
// Wave-level op, EXEC ignored, tracked by TENSORcnt (in-order per wave).
// ---------------------------------------------------------------------------
__device__ __forceinline__ void tdm_load_tile(unsigned int lds_off, const float* gptr) {
    unsigned long long ga = (unsigned long long)(uintptr_t)gptr;
    u32x4 g0;
    g0[0] = 1u;                                   // count=1 (valid), user mode
    g0[1] = lds_off;                              // lds_addr (bytes)
    g0[2] = (unsigned int)ga;                     // global_addr[31:0]
    g0[3] = ((unsigned int)(ga >> 32) & 0x01FFFFFFu) | (2u << 30); // ga[56:32] | type=2
    u32x8 g1;
    g1[0] = 2u << 16;                             // workgroup_mask=0, data_size=2 (4B)
    g1[1] = (unsigned int)DD << 16;               // tensor_dim0[15:0]=1024 (in [31:16])
    g1[2] = (unsigned int)TT << 16;               // dim0[31:16]=0 | tensor_dim1[15:0]=2048
    g1[3] = 64u << 16;                            // dim1[31:16]=0 | tile_dim0=64
    g1[4] = (unsigned int)KT;                     // tile_dim1=KT, tile_dim2=0
    g1[5] = (unsigned int)DD;                     // tensor_dim0_stride=1024 (low 32)
    g1[6] = 0u;                                   // stride[47:32]=0 | dim1_stride lo=0
    g1[7] = 0u;
    asm volatile("tensor_load_to_lds %0, %1" :: "s"(g0), "s"(g1) : "memory");
}

// ---------------------------------------------------------------------------
// Kernel 1: per-batch histogram of step ids -> counts[b][s] (as float)
// ---------------------------------------------------------------------------
__global__ __launch_bounds__(256) void counts_kernel(const int* __restrict__ sid,
                                                     float* __restrict__ cnts) {
    __shared__ int c[SS];
    const int b = blockIdx.x;
    if (threadIdx.x < SS) c[threadIdx.x] = 0;
    __syncthreads();
    for (int t = threadIdx.x; t < TT; t += 256) {
        int s = sid[b * TT + t];            // values 1..32
        atomicAdd(&c[s - 1], 1);
    }
    __syncthreads();
    if (threadIdx.x < SS) cnts[b * SS + threadIdx.x] = (float)c[threadIdx.x];
}

// ---------------------------------------------------------------------------
// Kernel 2: H_sum[b,s,d] = sum_t (step_id[b,t]==s+1) * |x[b,t,d]|
// GEMM: (S x T) onehot^T  @  (T x D) |x|   via V_WMMA_F32_16X16X4_F32 (exact)
//
// Grid: 16 batches x 16 d-slabs of 64 columns. Block = 256 thr = 8 waves.
// TDM double-buffers 128x64 |x| tiles HBM->LDS while waves run WMMA.
//
// WMMA f32 16x16x4 fragment layouts (ISA 7.12.2):
//   A (16x4): lane l row M=l%16; VGPR0 holds K=0 (lanes 0-15) / K=2 (16-31),
//             VGPR1 holds K=1 / K=3.  (onehot, built in registers)
//   B (4x16): lane l col N=l%16; same K split across lane halves.
//   C/D (16x16): VGPR v -> row M=v (lanes 0-15) / M=v+8 (lanes 16-31), N=l%16.
// ---------------------------------------------------------------------------
__global__ __launch_bounds__(256) void h_gemm_kernel(const float* __restrict__ x,
                                                     const int* __restrict__ sid,
                                                     float* __restrict__ Hs) {
    __shared__ int   sid_lds[TT];            // 8 KB: step ids for this batch
    __shared__ float x_lds[2][KT * 64];      // 2 x 32 KB: raw x chunks (TDM dest)

    const int b    = blockIdx.x >> 4;
    const int d0   = (blockIdx.x & 15) << 6;     // 64-column slab
    const int tid  = threadIdx.x;
    const int lane = tid & 31;
    const int w    = tid >> 5;
    const int mt   = w & 1;                      // s-tile (0 or 1)
    const int nt   = w >> 1;                     // d-tile within slab (0..3)

    // Stage all step ids for this batch once (covered by first barrier).
    for (int i = tid; i < TT; i += 256) sid_lds[i] = sid[b * TT + i];

    const int sval  = mt * 16 + (lane & 15) + 1; // step value this lane matches
    const int khalf = (lane >> 4) << 1;          // 0 (lanes 0-15) or 2 (16-31)
    const int ncol  = nt * 16 + (lane & 15);     // column within 64-wide slab

    const float* xbase = x + (size_t)b * TT * DD + d0;

    // Prologue: wave 0 kicks TDM for chunk 0 into buffer 0.
    if (w == 0)
        tdm_load_tile((unsigned int)(uintptr_t)&x_lds[0][0], xbase);

    v8f acc = {};                                // 16x16 f32 accumulator

    for (int c = 0; c < NC; ++c) {
        const int buf = c & 1;
        if (c + 1 < NC) {
            if (w == 0) {
                // Stream next chunk into the other buffer, then wait for
                // chunk c (TDM ops complete in order: outstanding<=1 => c done).
                tdm_load_tile((unsigned int)(uintptr_t)&x_lds[buf ^ 1][0],
                              xbase + (size_t)(c + 1) * KT * DD);
                __builtin_amdgcn_s_wait_tensorcnt(1);
            }
        } else {
            if (w == 0) __builtin_amdgcn_s_wait_tensorcnt(0);
        }
        __syncthreads();     // chunk c visible to all waves

        const float* xc = &x_lds[buf][0];
        const int*   sc = &sid_lds[c * KT];
        #pragma unroll 8
        for (int k0 = 0; k0 < KT; k0 += 4) {
            // A fragment: onehot built in registers from step ids
            int kg = k0 + khalf;
            int s0 = sc[kg];
            int s1 = sc[kg + 1];
            v2f af;
            af.x = (s0 == sval) ? 1.0f : 0.0f;
            af.y = (s1 == sval) ? 1.0f : 0.0f;
            // B fragment: |x| columns from LDS (abs applied here)
            v2f bf;
            bf.x = fabsf(xc[kg * 64 + ncol]);
            bf.y = fabsf(xc[(kg + 1) * 64 + ncol]);
            // D = A(16x4) * B(4x16) + C   (exact fp32)
            acc = __builtin_amdgcn_wmma_f32_16x16x4_f32(
                false, af, false, bf, (short)0, acc, false, false);
        }
        __syncthreads();     // all reads of buf done before TDM re-targets it
    }

    // Store 16x16 tile of H_sum
    const int srow0 = mt * 16 + ((lane >> 4) << 3);   // +0 or +8 per C/D layout
    #pragma unroll
    for (int v = 0; v < 8; ++v) {
        int s = srow0 + v;
        Hs[((size_t)(b * SS + s)) * DD + d0 + ncol] = acc[v];
    }
}

// ---------------------------------------------------------------------------
// Kernel 3: E[b,s'] = sum_d relu(Hs[b,s',d]/c0 - Hs[b,s'+1,d]/c1)^2
// ---------------------------------------------------------------------------
__global__ __launch_bounds__(256) void e_kernel(const float* __restrict__ Hs,
                                                const float* __restrict__ cnts,
                                                float* __restrict__ E) {
    __shared__ float red[256];
    const int blk = blockIdx.x;          // 0..495
    const int b   = blk / (SS - 1);
    const int sp  = blk % (SS - 1);
    const float c0 = cnts[b * SS + sp];
    const float c1 = cnts[b * SS + sp + 1];
    const float* h0 = Hs + ((size_t)(b * SS + sp)) * DD;
    const float* h1 = Hs + ((size_t)(b * SS + sp + 1)) * DD;
    float partial = 0.0f;
    for (int d = threadIdx.x; d < DD; d += 256) {
        float diff = h0[d] / c0 - h1[d] / c1;
        float r = diff > 0.0f ? diff : 0.0f;
        partial += r * r;
    }
    red[threadIdx.x] = partial;
    __syncthreads();
    for (int off = 128; off > 0; off >>= 1) {
        if (threadIdx.x < off) red[threadIdx.x] += red[threadIdx.x + off];
        __syncthreads();
    }
    if (threadIdx.x == 0) E[blk] = red[0];
}

// ---------------------------------------------------------------------------
// Kernel 4: margin loss + final scalar
// ---------------------------------------------------------------------------
__global__ __launch_bounds__(32) void final_kernel(const float* __restrict__ E,
                                                   const int* __restrict__ labels,
                                                   float* __restrict__ out) {
    __shared__ float red[BB];
    const int t = threadIdx.x;
    if (t < BB) {
        float pos = 0.0f, neg = 0.0f;
        for (int s = 0; s < SS - 1; ++s) {
            float e = E[t * (SS - 1) + s];
            pos += e;
            float r = ALPHA_C - e;
            neg += (r > 0.0f ? r : 0.0f);
        }
        pos /= (float)(SS - 1);
        neg /= (float)(SS - 1);
        red[t] = (labels[t] == 1) ? pos : neg;
    }
    __syncthreads();
    if (t == 0) {
        float s = 0.0f;
        for (int i = 0; i < BB; ++i) s += red[i];
        out[0] = s / ((float)BB + 1e-9f);
    }
}

// ---------------------------------------------------------------------------
extern "C" void kernel_launch(void* const* d_in, const int* in_sizes, int n_in,
                              void* d_out, int out_size, void* d_ws, size_t ws_size,
                              hipStream_t stream) {
    const float* x      = (const float*)d_in[0];   // (B,T,D) fp32
    const int*   sid    = (const int*)d_in[1];     // (B,T) int
    const int*   labels = (const int*)d_in[2];     // (B,) int
    float* out = (float*)d_out;

    // Workspace layout (floats): H_sum | counts | E
    float* Hs   = (float*)d_ws;                    // B*S*D = 524288
    float* cnts = Hs + (size_t)BB * SS * DD;       // B*S   = 512
    float* E    = cnts + BB * SS;                  // B*(S-1) = 496

    counts_kernel<<<BB, 256, 0, stream>>>(sid, cnts);
    h_gemm_kernel<<<BB * (DD / 64), 256, 0, stream>>>(x, sid, Hs);
    e_kernel<<<BB * (SS - 1), 256, 0, stream>>>(Hs, cnts, E);
    final_kernel<<<1, 32, 0, stream>>>(E, labels, out);
}